// PatchLoss_9225589752592
// MI455X (gfx1250) — compile-verified
//
#include <hip/hip_runtime.h>

// PatchLoss on MI455X (gfx1250, wave32).
// Memory-bound: 256 MiB of f32 reads -> ~11us floor at 23.3 TB/s.
// Strategy: one 256-thread workgroup per (batch, patch-row) strip (16x1024),
// coalesced b128 global loads -> |a-b| -> padded LDS, then per-patch 16x16
// sums via V_WMMA_F32_16X16X4_F32 with a ones B-matrix (D[m,n] = sum_k A[m,k]).

typedef __attribute__((ext_vector_type(2))) float v2f;
typedef __attribute__((ext_vector_type(8))) float v8f;

#define BATCH 32
#define HDIM  1024
#define WDIM  1024
#define PSZ   16
#define NH    (HDIM / PSZ)   // 64 patch-rows
#define NW    (WDIM / PSZ)   // 64 patches per strip
// Padded LDS row stride (floats). 1028 mod 64 == 4 -> the WMMA A-fragment
// reads (lane m at word m*1028 + t*16 + 4c + 2*half, 2 words each) land on
// 32 distinct banks; staging float4 stores stay 16B-aligned.
#define LSTRIDE 1028

__global__ __launch_bounds__(256) void PatchLoss_strip_kernel(
    const float* __restrict__ out, const float* __restrict__ tgt,
    unsigned int* __restrict__ smax) {
  __shared__ float lds[PSZ * LSTRIDE];   // 16 * 1028 * 4B = 65,792 B
  __shared__ float wmaxs[8];

  const int tid = threadIdx.x;
  const int blk = blockIdx.x;            // b * NH + ph
  const int b   = blk / NH;
  const int ph  = blk - b * NH;

  // ---- Stage |out - tgt| for the 16-row strip into padded LDS -------------
  // Thread tid owns columns [4*tid, 4*tid+3] of every row: fully coalesced
  // global_load_b128 pairs, ds_store_b128 into the padded layout.
  const size_t rowbase = ((size_t)b * HDIM + (size_t)ph * PSZ) * (size_t)WDIM;
  #pragma unroll 4
  for (int j = 0; j < PSZ; ++j) {
    const size_t g = rowbase + (size_t)j * WDIM + (size_t)(tid * 4);
    const float4 a4 = *(const float4*)(out + g);
    const float4 t4 = *(const float4*)(tgt + g);
    float4 d4;
    d4.x = fabsf(a4.x - t4.x);
    d4.y = fabsf(a4.y - t4.y);
    d4.z = fabsf(a4.z - t4.z);
    d4.w = fabsf(a4.w - t4.w);
    *(float4*)(&lds[j * LSTRIDE + tid * 4]) = d4;
  }
  __syncthreads();

  // ---- Per-patch sums via WMMA: D = A(16x4) * ones(4x16) + C --------------
  // A-fragment f32 16x4 layout (ISA 7.12.2): lanes 0-15 hold M=lane with
  // {K=0,K=1} in VGPR{0,1}; lanes 16-31 hold {K=2,K=3}. After 4 K-chunks the
  // accumulator holds row sums replicated across N.
  const int lane = tid & 31;
  const int wave = tid >> 5;
  const int m    = lane & 15;
  const int half = lane >> 4;

  v2f ones; ones[0] = 1.0f; ones[1] = 1.0f;

  float wmax = 0.0f;
  for (int i = 0; i < 8; ++i) {               // 8 patches per wave, 64/strip
    const int t = wave * 8 + i;
    const float* abase = &lds[m * LSTRIDE + t * PSZ + 2 * half];
    v8f acc = {0.f, 0.f, 0.f, 0.f, 0.f, 0.f, 0.f, 0.f};
    #pragma unroll
    for (int c = 0; c < 4; ++c) {
      const v2f av = *(const v2f*)(abase + 4 * c);   // 8B-aligned ds_load_b64
      acc = __builtin_amdgcn_wmma_f32_16x16x4_f32(
          /*neg_a=*/false, av, /*neg_b=*/false, ones,
          /*c_mod=*/(short)0, acc, /*reuse_a=*/false, /*reuse_b=*/false);
    }
    // C/D layout: VGPR j = rows {j | j+8}; every N column identical.
    float s = acc[0] + acc[1] + acc[2] + acc[3] +
              acc[4] + acc[5] + acc[6] + acc[7];   // rows 0-7 | rows 8-15
    s += __shfl_xor(s, 16, 32);                    // full 16x16 patch sum
    wmax = fmaxf(wmax, s);
  }

  if (lane == 0) wmaxs[wave] = wmax;
  __syncthreads();
  if (tid == 0) {
    float v = wmaxs[0];
    #pragma unroll
    for (int w = 1; w < 8; ++w) v = fmaxf(v, wmaxs[w]);
    // Non-negative floats: uint compare == IEEE compare.
    atomicMax(&smax[b], __float_as_uint(v));
  }
}

__global__ void PatchLoss_init_kernel(unsigned int* smax) {
  if (threadIdx.x < BATCH) smax[threadIdx.x] = 0u;
}

__global__ void PatchLoss_final_kernel(const unsigned int* __restrict__ smax,
                                       float* __restrict__ o) {
  if (threadIdx.x == 0) {
    float acc = 0.0f;
    for (int b = 0; b < BATCH; ++b) acc += __uint_as_float(smax[b]);
    o[0] = acc / ((float)BATCH * (float)(PSZ * PSZ));  // /256 patch mean, /32 batch mean
  }
}

extern "C" void kernel_launch(void* const* d_in, const int* in_sizes, int n_in,
                              void* d_out, int out_size, void* d_ws, size_t ws_size,
                              hipStream_t stream) {
  const float* out_t = (const float*)d_in[0];
  const float* tgt_t = (const float*)d_in[1];
  unsigned int* smax = (unsigned int*)d_ws;   // 32 uint accumulators
  float* o = (float*)d_out;

  PatchLoss_init_kernel<<<1, 32, 0, stream>>>(smax);
  PatchLoss_strip_kernel<<<BATCH * NH, 256, 0, stream>>>(out_t, tgt_t, smax);
  PatchLoss_final_kernel<<<1, 32, 0, stream>>>(smax, o);
}